// TimeAwareScaledDotProduct_43550968381458
// MI455X (gfx1250) — compile-verified
//
#include <hip/hip_runtime.h>
#include <math.h>

typedef float v2f __attribute__((ext_vector_type(2)));
typedef float v4f __attribute__((ext_vector_type(4)));
typedef float v8f __attribute__((ext_vector_type(8)));

#define B_  4
#define H_  8
#define L_  256
#define D_  64
#define QT  16
#define MASKV (-4294967295.0f)

__global__ __launch_bounds__(256, 1)
void ta_sdpa_kernel(const float* __restrict__ Q, const float* __restrict__ K,
                    const float* __restrict__ V, const float* __restrict__ TK,
                    const float* __restrict__ TV, const float* __restrict__ AM,
                    const unsigned char* __restrict__ PM,
                    float* __restrict__ Oout, float* __restrict__ Wout)
{
    // Padded LDS tiles (row pad 68/260 -> conflict-free column access, 16B-aligned rows)
    __shared__ float Qs [QT][68];     //  4,352 B
    __shared__ float KVs[L_][68];     // 69,632 B (K, then reused for V)
    __shared__ float Es [QT][260];    // 16,640 B (energy -> weights)
    __shared__ float Os [QT][D_];     //  4,096 B

    const int t    = threadIdx.x;
    const int wave = t >> 5;
    const int lane = t & 31;
    const int half = lane >> 4;   // which 16-lane half of the wave
    const int lid  = lane & 15;

    const int qt    = blockIdx.x;
    const int hh    = blockIdx.y;
    const int bb    = blockIdx.z;
    const int bh    = bb * H_ + hh;
    const int qbase = qt * QT;

    const size_t kvbase = (size_t)bh * L_ * D_;

    // ---------------- load Q tile + K into LDS (reused data: regular TH) ----------------
    {
        const v4f* gq = (const v4f*)(Q + kvbase + (size_t)qbase * D_);
        int row = t >> 4, c4 = t & 15;
        *(v4f*)&Qs[row][c4 * 4] = gq[t];

        const v4f* gk = (const v4f*)(K + kvbase);
        #pragma unroll
        for (int i = 0; i < 16; ++i) {
            int e4 = i * 256 + t;            // 4096 float4 total
            int r = e4 >> 4, c = e4 & 15;
            *(v4f*)&KVs[r][c * 4] = gk[e4];
        }
    }
    __syncthreads();

    // ---------------- E = Q * K^T via FP32 WMMA (16x16x4) ----------------
    // A: 16x4 f32  lane<16: M=lane, K={0,1}; lane>=16: M=lane-16, K={2,3}
    // B:  4x16 f32 lane<16: N=lane, K={0,1}; lane>=16: N=lane-16, K={2,3}
    // D: vgpr r -> rows r (lanes 0-15) and r+8 (lanes 16-31)
    for (int tt = 0; tt < 2; ++tt) {
        const int k0 = (wave * 2 + tt) * 16;   // 16 key-column tiles over 8 waves
        v8f acc = {};
        #pragma unroll
        for (int s = 0; s < 16; ++s) {         // K-dim d=64 in steps of 4
            const int dd = 4 * s + 2 * half;
            v2f a, b;
            a.x = Qs[lid][dd];        a.y = Qs[lid][dd + 1];
            b.x = KVs[k0 + lid][dd];  b.y = KVs[k0 + lid][dd + 1];
            acc = __builtin_amdgcn_wmma_f32_16x16x4_f32(false, a, false, b,
                                                        (short)0, acc, false, false);
        }
        #pragma unroll
        for (int r = 0; r < 8; ++r)
            Es[r + 8 * half][k0 + lid] = acc[r];
    }
    __syncthreads();

    // ------- relative-time key score: E[q][t] += dot(TK[q,t,:], Q[q,:]) -------
    // TK is streamed exactly once -> non-temporal loads (don't thrash L2)
    {
        const float* tkp = TK + ((((size_t)bh * L_ + qbase) * L_ + t) * D_);
        for (int q = 0; q < QT; ++q) {
            const v4f* p = (const v4f*)(tkp + (size_t)q * L_ * D_);
            float a0 = 0.f, a1 = 0.f, a2 = 0.f, a3 = 0.f;
            #pragma unroll
            for (int i = 0; i < 16; ++i) {
                v4f v = __builtin_nontemporal_load(p + i);
                a0 += v.x * Qs[q][4 * i + 0];
                a1 += v.y * Qs[q][4 * i + 1];
                a2 += v.z * Qs[q][4 * i + 2];
                a3 += v.w * Qs[q][4 * i + 3];
            }
            Es[q][t] += (a0 + a1) + (a2 + a3);
        }
    }
    __syncthreads();

    // ---------------- softmax (scale, masks), emit weights ----------------
    {
        const int r = t >> 4, j = t & 15;        // 16 lanes per query row
        const int qg = qbase + r;
        float ev[16];
        float mx = -INFINITY;
        #pragma unroll
        for (int i = 0; i < 16; ++i) {
            const int c = j + 16 * i;
            float e = Es[r][c] * 0.125f + AM[(size_t)qg * L_ + c];
            if (PM[bb * L_ + c]) e = MASKV;
            ev[i] = e;
            mx = fmaxf(mx, e);
        }
        #pragma unroll
        for (int m = 8; m; m >>= 1) mx = fmaxf(mx, __shfl_xor(mx, m, 16));
        float sum = 0.f;
        #pragma unroll
        for (int i = 0; i < 16; ++i) { ev[i] = __expf(ev[i] - mx); sum += ev[i]; }
        #pragma unroll
        for (int m = 8; m; m >>= 1) sum += __shfl_xor(sum, m, 16);
        const float rinv = 1.0f / sum;
        float* wrow = Wout + ((size_t)bh * L_ + qg) * L_;
        #pragma unroll
        for (int i = 0; i < 16; ++i) {
            const int c = j + 16 * i;
            const float w = ev[i] * rinv;
            Es[r][c] = w;
            __builtin_nontemporal_store(w, wrow + c);   // write-once output
        }
    }
    __syncthreads();

    // ---------------- reload V into the K buffer ----------------
    {
        const v4f* gv = (const v4f*)(V + kvbase);
        #pragma unroll
        for (int i = 0; i < 16; ++i) {
            int e4 = i * 256 + t;
            int r = e4 >> 4, c = e4 & 15;
            *(v4f*)&KVs[r][c * 4] = gv[e4];
        }
    }
    __syncthreads();

    // ---------------- O = W * V via FP32 WMMA ----------------
    if (wave < 4) {
        const int d0 = wave * 16;
        v8f acc = {};
        for (int s = 0; s < 64; ++s) {          // K-dim k=256 in steps of 4
            const int kk = 4 * s + 2 * half;
            v2f a, b;
            a.x = Es[lid][kk];            a.y = Es[lid][kk + 1];
            b.x = KVs[kk][d0 + lid];      b.y = KVs[kk + 1][d0 + lid];
            acc = __builtin_amdgcn_wmma_f32_16x16x4_f32(false, a, false, b,
                                                        (short)0, acc, false, false);
        }
        #pragma unroll
        for (int r = 0; r < 8; ++r)
            Os[r + 8 * half][d0 + lid] = acc[r];
    }
    __syncthreads();

    // ------- relative-time value term: O[q][d] += sum_k w[q][k]*TV[q,k,d] -------
    // TV is streamed exactly once -> non-temporal loads
    {
        const int d  = t & 63;
        const int kc = t >> 6;                   // 4 chunks of 64 keys
        const float* tvp = TV + ((((size_t)bh * L_ + qbase) * L_ + kc * 64) * D_) + d;
        for (int q = 0; q < QT; ++q) {
            const float* pq = tvp + (size_t)q * L_ * D_;
            float acc = 0.f;
            #pragma unroll 8
            for (int kk = 0; kk < 64; ++kk)
                acc += Es[q][kc * 64 + kk] *
                       __builtin_nontemporal_load(pq + (size_t)kk * D_);
            atomicAdd(&Os[q][d], acc);
        }
    }
    __syncthreads();

    // ---------------- store O (write-once output) ----------------
    {
        const int row = t >> 4, c4 = t & 15;
        v4f v = *(const v4f*)&Os[row][c4 * 4];
        __builtin_nontemporal_store(
            v, (v4f*)&Oout[((size_t)bh * L_ + qbase + row) * D_ + c4 * 4]);
    }
}

extern "C" void kernel_launch(void* const* d_in, const int* in_sizes, int n_in,
                              void* d_out, int out_size, void* d_ws, size_t ws_size,
                              hipStream_t stream) {
    (void)in_sizes; (void)n_in; (void)d_ws; (void)ws_size; (void)out_size;
    const float*         Q  = (const float*)d_in[0];
    const float*         K  = (const float*)d_in[1];
    const float*         V  = (const float*)d_in[2];
    const float*         TK = (const float*)d_in[3];
    const float*         TV = (const float*)d_in[4];
    const float*         AM = (const float*)d_in[5];
    const unsigned char* PM = (const unsigned char*)d_in[6];

    float* Oout = (float*)d_out;                          // (b,h,L,d)
    float* Wout = Oout + (size_t)B_ * H_ * L_ * D_;       // (b,h,L,L)

    dim3 grid(L_ / QT, H_, B_);
    dim3 block(256);
    ta_sdpa_kernel<<<grid, block, 0, stream>>>(Q, K, V, TK, TV, AM, PM, Oout, Wout);
}